// VariationalLinear_15539191677708
// MI455X (gfx1250) — compile-verified
//
#include <hip/hip_runtime.h>

// ---- vector types (clang ext vectors: safe for __builtin_nontemporal_load) ----
typedef __attribute__((ext_vector_type(2))) float v2f;
typedef __attribute__((ext_vector_type(4))) float v4f;
typedef __attribute__((ext_vector_type(8))) float v8f;

#define IN_SIZE 4096
#define OUT_SIZE 8192
#define ROWS_PER_BLOCK 16
#define WAVES_PER_BLOCK 8
#define THREADS_PER_BLOCK 256
#define K_PER_WAVE (IN_SIZE / WAVES_PER_BLOCK) /* 512 floats */

// Numerically stable softplus: max(z,0) + log(1 + exp(-|z|))
__device__ __forceinline__ float softplusf(float z) {
    return fmaxf(z, 0.0f) + __logf(1.0f + __expf(-fabsf(z)));
}

__global__ __launch_bounds__(THREADS_PER_BLOCK)
void varlinear_kernel(const float* __restrict__ x,
                      const float* __restrict__ mu,
                      const float* __restrict__ rho,
                      const float* __restrict__ bias_mu,
                      const float* __restrict__ bias_rho,
                      const float* __restrict__ eps_w,
                      const float* __restrict__ eps_b,
                      float* __restrict__ out) {
    __shared__ float lds[THREADS_PER_BLOCK * ROWS_PER_BLOCK]; // 16 KB

    const int lane  = threadIdx.x & 31;
    const int wave  = threadIdx.x >> 5;
    const int rbase = blockIdx.x * ROWS_PER_BLOCK;

    const v4f* __restrict__ x4   = (const v4f*)x;
    const int  kq_wave = wave * (K_PER_WAVE / 4); // float4 base index of this wave's K-slice

    // Preload this lane's x slice once (reused by all 16 rows).
    v4f xv[4];
    #pragma unroll
    for (int c = 0; c < 4; ++c) xv[c] = x4[kq_wave + c * 32 + lane];

    // Per-row streaming pointers (bumped by one row = 1024 float4 per iteration).
    const long base = (long)rbase * (IN_SIZE / 4) + kq_wave + lane;
    const v4f* __restrict__ mp = (const v4f*)mu    + base;
    const v4f* __restrict__ rp = (const v4f*)rho   + base;
    const v4f* __restrict__ ep = (const v4f*)eps_w + base;

    // ---- bandwidth-bound main loop ----
    // Row loop stays rolled (bounded VGPR pressure: 12 b128 loads in flight);
    // K-chunk loop unrolled so all 12 loads issue back-to-back per row.
    #pragma unroll 1
    for (int r = 0; r < ROWS_PER_BLOCK; ++r) {
        v4f a4 = {0.0f, 0.0f, 0.0f, 0.0f};
        #pragma unroll
        for (int c = 0; c < 4; ++c) {
            const v4f m  = __builtin_nontemporal_load(&mp[c * 32]);
            const v4f rh = __builtin_nontemporal_load(&rp[c * 32]);
            const v4f e  = __builtin_nontemporal_load(&ep[c * 32]);
            float w;
            w = fmaf(softplusf(rh.x), e.x, m.x); a4.x = fmaf(w, xv[c].x, a4.x);
            w = fmaf(softplusf(rh.y), e.y, m.y); a4.y = fmaf(w, xv[c].y, a4.y);
            w = fmaf(softplusf(rh.z), e.z, m.z); a4.z = fmaf(w, xv[c].z, a4.z);
            w = fmaf(softplusf(rh.w), e.w, m.w); a4.w = fmaf(w, xv[c].w, a4.w);
        }
        // Deposit this row's partial directly: lds[p*16 + r], p = lane index in block.
        lds[threadIdx.x * ROWS_PER_BLOCK + r] = (a4.x + a4.y) + (a4.z + a4.w);
        mp += IN_SIZE / 4;
        rp += IN_SIZE / 4;
        ep += IN_SIZE / 4;
    }
    __syncthreads();

    // Only wave 0 continues: guarantees EXEC == all-ones in the WMMA region.
    if (threadIdx.x >= 32) return;

    const int L = threadIdx.x;
    const int r = L & 15;   // row within tile
    const int h = L >> 4;   // lane half

    // Reduce 256 partials/row -> 4 group sums/row, placed directly in the
    // WMMA 16x4 f32 A-fragment layout: lanes 0-15 hold K=0 (v0) / K=1 (v1),
    // lanes 16-31 hold K=2 (v0) / K=3 (v1), M = lane % 16.
    float a0 = 0.0f, a1 = 0.0f;
    const int g0 = 128 * h;
    for (int q = 0; q < 64; ++q) a0 += lds[(g0 + q)      * ROWS_PER_BLOCK + r];
    for (int q = 0; q < 64; ++q) a1 += lds[(g0 + 64 + q) * ROWS_PER_BLOCK + r];

#if __has_builtin(__builtin_amdgcn_wmma_f32_16x16x4_f32)
    // C fragment = bias, matching the 16x16 f32 C/D layout:
    // VGPR rr: lanes 0-15 -> M=rr, lanes 16-31 -> M=rr+8.
    v8f cfrag;
    #pragma unroll
    for (int rr = 0; rr < 8; ++rr) {
        const int row = rbase + 8 * h + rr;
        cfrag[rr] = fmaf(softplusf(bias_rho[row]), eps_b[row], bias_mu[row]);
    }

    v2f a; a[0] = a0;   a[1] = a1;
    v2f b; b[0] = 1.0f; b[1] = 1.0f;   // B[k][n] = 1 -> D[m][n] = bias[m] + sum_k A[m][k]
    v8f d = __builtin_amdgcn_wmma_f32_16x16x4_f32(
        /*neg_a=*/false, a, /*neg_b=*/false, b,
        /*c_mod=*/(short)0, cfrag, /*reuse_a=*/false, /*reuse_b=*/false);

    // Every column of D is identical; lane 0 (rows 0-7) and lane 16 (rows 8-15) store.
    if (r == 0) {
        #pragma unroll
        for (int rr = 0; rr < 8; ++rr)
            out[rbase + 8 * h + rr] = d[rr];
    }
#else
    // Fallback: shuffle across lane halves, then bias + store from lanes 0-15.
    float s = a0 + a1;
    float t = s + __shfl_xor(s, 16, 32);
    if (h == 0) {
        const int row = rbase + r;
        out[row] = t + fmaf(softplusf(bias_rho[row]), eps_b[row], bias_mu[row]);
    }
#endif
}

extern "C" void kernel_launch(void* const* d_in, const int* in_sizes, int n_in,
                              void* d_out, int out_size, void* d_ws, size_t ws_size,
                              hipStream_t stream) {
    const float* x        = (const float*)d_in[0];
    const float* mu       = (const float*)d_in[1];
    const float* rho      = (const float*)d_in[2];
    const float* bias_mu  = (const float*)d_in[3];
    const float* bias_rho = (const float*)d_in[4];
    const float* eps_w    = (const float*)d_in[5];
    const float* eps_b    = (const float*)d_in[6];
    float* out = (float*)d_out;

    dim3 grid(OUT_SIZE / ROWS_PER_BLOCK); // 512 blocks
    dim3 block(THREADS_PER_BLOCK);        // 8 wave32 waves
    varlinear_kernel<<<grid, block, 0, stream>>>(x, mu, rho, bias_mu, bias_rho,
                                                 eps_w, eps_b, out);
}